// GroupedQueryAttention_64484638982880
// MI455X (gfx1250) — compile-verified
//
#include <hip/hip_runtime.h>
#include <math.h>

typedef __attribute__((ext_vector_type(2))) float v2f;
typedef __attribute__((ext_vector_type(8))) float v8f;

#define NEG_BIG (-1e30f)

// ---------------------------------------------------------------------------
// Generic fp32 WMMA GEMM:  C[m][n] = sum_k A[m][k] * W[n][k]
// A: (M,K) row-major, W: (N,K) row-major, C: (M,N) row-major.
// One wave computes a 64x64 tile; K stepped by 4 using v_wmma_f32_16x16x4_f32.
// Fragment layouts (ISA 7.12.2, 32-bit):
//   A 16x4:  lane L holds A[M = L%16][K = 2*(L/16)+j] in component j
//   B 4x16:  lane L holds B[K = 2*(L/16)+j][N = L%16]  -> W[N][K], vectorizable
//   C 16x16: VGPR r, lanes 0-15 -> M=r, lanes 16-31 -> M=r+8; N = L%16
// ---------------------------------------------------------------------------
__global__ __launch_bounds__(32)
void gemm_nt_f32_wmma(const float* __restrict__ A, const float* __restrict__ W,
                      float* __restrict__ C, int M, int N, int K) {
    const int lane = threadIdx.x;
    const int half = lane >> 4;
    const int l16  = lane & 15;
    const int mb = blockIdx.y * 64;
    const int nb = blockIdx.x * 64;

    v8f acc[4][4];
    const v8f vzero = {0.f, 0.f, 0.f, 0.f, 0.f, 0.f, 0.f, 0.f};
#pragma unroll
    for (int i = 0; i < 4; ++i)
#pragma unroll
        for (int j = 0; j < 4; ++j) acc[i][j] = vzero;

    // per-lane base pointers (row fixed per fragment index, K varies)
    const float* arow[4];
    const float* wrow[4];
#pragma unroll
    for (int i = 0; i < 4; ++i) {
        arow[i] = A + (size_t)(mb + i * 16 + l16) * K + 2 * half;
        wrow[i] = W + (size_t)(nb + i * 16 + l16) * K + 2 * half;
    }

    for (int k = 0; k < K; k += 4) {
        v2f a[4], b[4];
#pragma unroll
        for (int i = 0; i < 4; ++i) {
            a[i] = *(const v2f*)(arow[i] + k);
            b[i] = *(const v2f*)(wrow[i] + k);
        }
#pragma unroll
        for (int i = 0; i < 4; ++i)
#pragma unroll
            for (int j = 0; j < 4; ++j)
                acc[i][j] = __builtin_amdgcn_wmma_f32_16x16x4_f32(
                    false, a[i], false, b[j], (short)0, acc[i][j], false, false);
    }

#pragma unroll
    for (int i = 0; i < 4; ++i)
#pragma unroll
        for (int j = 0; j < 4; ++j)
#pragma unroll
            for (int r = 0; r < 8; ++r) {
                const int row = mb + i * 16 + r + 8 * half;
                const int col = nb + j * 16 + l16;
                C[(size_t)row * N + col] = acc[i][j][r];
            }
}

// ---------------------------------------------------------------------------
// RoPE, interleaved-pair convention: for pair p (elements 2p, 2p+1) of a
// 64-wide head at sequence position s:
//   theta = s * 10000^(-2p/64);  (x1,x2) -> (x1*c - x2*s, x1*s + x2*c)
// X laid out (B, S, H, 64) row-major; idx enumerates (b,s,h,p), p fastest,
// so the pair lives at elements 2*idx, 2*idx+1.
// ---------------------------------------------------------------------------
__global__ __launch_bounds__(256)
void rope_kernel(float* __restrict__ X, int S, int H, int total_pairs) {
    const int idx = blockIdx.x * blockDim.x + threadIdx.x;
    if (idx >= total_pairs) return;
    const int p = idx & 31;                 // D_HEAD/2 = 32 pairs
    const int s = (idx / (32 * H)) % S;
    const float inv = powf(10000.0f, -(float)(2 * p) * (1.0f / 64.0f));
    const float th = (float)s * inv;
    const float c = cosf(th), sn = sinf(th);
    float* v = X + (size_t)idx * 2;
    const float x1 = v[0], x2 = v[1];
    v[0] = x1 * c - x2 * sn;
    v[1] = x1 * sn + x2 * c;
}

// ---------------------------------------------------------------------------
// Flash-style causal GQA attention. One wave per (b, q-head h, 16-row block).
// Q,K,V laid out (B,S,H,64) row-major with H=32 for Q, H=8 for K/V.
// kv head = h/4 (reference groups q head kvh*G+g with kv head kvh, G=4).
// S = (Q @ K^T)/8 via 16 fp32 WMMAs; online softmax with __shfl_xor row
// reductions on the C-fragment layout; P transposed to A layout through a
// 1KB LDS tile (same-wave LDS ops are in-order -> no barrier needed);
// O += P @ V via 16 more WMMAs per key block.
// ---------------------------------------------------------------------------
__global__ __launch_bounds__(32)
void gqa_attn_wmma(const float* __restrict__ Q, const float* __restrict__ K,
                   const float* __restrict__ V, float* __restrict__ O, int S) {
    __shared__ float lds_p[256];            // 16x16 probability tile

    const int lane = threadIdx.x;
    const int half = lane >> 4;
    const int l16  = lane & 15;
    const int mb = blockIdx.x;              // query row block (16 rows)
    const int h  = blockIdx.y;              // q head
    const int b  = blockIdx.z;
    const int hk = h >> 2;                  // kv head
    const int mbase = mb * 16;

    // Persistent Q A-fragments: 16 k-steps covering Dh=64.
    v2f qa[16];
    {
        const float* qrow =
            Q + ((size_t)(b * S + mbase + l16) * 32 + h) * 64 + 2 * half;
#pragma unroll
        for (int t = 0; t < 16; ++t) qa[t] = *(const v2f*)(qrow + t * 4);
    }

    float m_i[8], l_i[8];
    v8f o[4];
    const v8f vzero = {0.f, 0.f, 0.f, 0.f, 0.f, 0.f, 0.f, 0.f};
#pragma unroll
    for (int r = 0; r < 8; ++r) { m_i[r] = NEG_BIG; l_i[r] = 0.f; }
#pragma unroll
    for (int nt = 0; nt < 4; ++nt) o[nt] = vzero;

    for (int kb = 0; kb <= mb; ++kb) {
        const int kbb = kb * 16;

        // ---- S = Q @ K^T (B-frag: lane = key l16, components = dims) ----
        v8f sfr = vzero;
        const float* krow =
            K + ((size_t)(b * S + kbb + l16) * 8 + hk) * 64 + 2 * half;
#pragma unroll
        for (int t = 0; t < 16; ++t) {
            const v2f kf = *(const v2f*)(krow + t * 4);
            sfr = __builtin_amdgcn_wmma_f32_16x16x4_f32(
                false, qa[t], false, kf, (short)0, sfr, false, false);
        }

        // ---- online softmax per row (row = r + 8*half, col = l16) ----
#pragma unroll
        for (int r = 0; r < 8; ++r) {
            float sv = sfr[r] * 0.125f;     // 1/sqrt(64)
            if (kb == mb) {
                const int rowg = r + 8 * half;
                if (l16 > rowg) sv = NEG_BIG;   // causal mask (diag block)
            }
            float v = sv;
            v = fmaxf(v, __shfl_xor(v, 1));
            v = fmaxf(v, __shfl_xor(v, 2));
            v = fmaxf(v, __shfl_xor(v, 4));
            v = fmaxf(v, __shfl_xor(v, 8));
            const float mnew = fmaxf(m_i[r], v);
            const float sc = expf(m_i[r] - mnew);
            const float p  = expf(sv - mnew);
            float rs = p;
            rs += __shfl_xor(rs, 1);
            rs += __shfl_xor(rs, 2);
            rs += __shfl_xor(rs, 4);
            rs += __shfl_xor(rs, 8);
            l_i[r] = l_i[r] * sc + rs;
            m_i[r] = mnew;
#pragma unroll
            for (int nt = 0; nt < 4; ++nt) o[nt][r] *= sc;
            lds_p[(r + 8 * half) * 16 + l16] = p;   // row-major P tile
        }

        // ---- O += P @ V (P reloaded from LDS in A layout) ----
#pragma unroll
        for (int t = 0; t < 4; ++t) {
            const v2f pa = *(const v2f*)(&lds_p[l16 * 16 + t * 4 + 2 * half]);
            const int k0 = kbb + t * 4 + 2 * half;
            const float* v0 = V + ((size_t)(b * S + k0) * 8 + hk) * 64 + l16;
            const float* v1 = V + ((size_t)(b * S + k0 + 1) * 8 + hk) * 64 + l16;
#pragma unroll
            for (int nt = 0; nt < 4; ++nt) {
                v2f vf;
                vf.x = v0[nt * 16];
                vf.y = v1[nt * 16];
                o[nt] = __builtin_amdgcn_wmma_f32_16x16x4_f32(
                    false, pa, false, vf, (short)0, o[nt], false, false);
            }
        }
    }

    // ---- normalize and write attn in (B,S,32,64) == (B,S,2048) layout ----
#pragma unroll
    for (int nt = 0; nt < 4; ++nt)
#pragma unroll
        for (int r = 0; r < 8; ++r) {
            const int srow = mbase + r + 8 * half;
            O[((size_t)(b * S + srow) * 32 + h) * 64 + nt * 16 + l16] =
                o[nt][r] / l_i[r];
        }
}

// ---------------------------------------------------------------------------
extern "C" void kernel_launch(void* const* d_in, const int* in_sizes, int n_in,
                              void* d_out, int out_size, void* d_ws, size_t ws_size,
                              hipStream_t stream) {
    const float* x  = (const float*)d_in[0];   // (B,S,2048)
    const float* Wq = (const float*)d_in[1];   // (2048,2048)
    const float* Wk = (const float*)d_in[2];   // (512,2048)
    const float* Wv = (const float*)d_in[3];   // (512,2048)
    const float* Wo = (const float*)d_in[4];   // (2048,2048)
    float* out = (float*)d_out;                // (B,S,2048)

    const int B = 2, S = 2048, D = 2048;
    const int M = B * S;                       // 4096 token rows

    // Workspace carve-up (80 MB total): Q | K | V | attn
    float* Qb = (float*)d_ws;                  // M*2048
    float* Kb = Qb + (size_t)M * 2048;         // M*512
    float* Vb = Kb + (size_t)M * 512;          // M*512
    float* Ab = Vb + (size_t)M * 512;          // M*2048

    const dim3 wave(32);

    // QKV projections: C = x @ W^T
    gemm_nt_f32_wmma<<<dim3(2048 / 64, M / 64), wave, 0, stream>>>(x, Wq, Qb, M, 2048, D);
    gemm_nt_f32_wmma<<<dim3(512 / 64,  M / 64), wave, 0, stream>>>(x, Wk, Kb, M, 512, D);
    gemm_nt_f32_wmma<<<dim3(512 / 64,  M / 64), wave, 0, stream>>>(x, Wv, Vb, M, 512, D);

    // RoPE on Q (H=32) and K (H=8)
    const int qpairs = M * 32 * 32;
    const int kpairs = M * 8 * 32;
    rope_kernel<<<(qpairs + 255) / 256, 256, 0, stream>>>(Qb, S, 32, qpairs);
    rope_kernel<<<(kpairs + 255) / 256, 256, 0, stream>>>(Kb, S, 8, kpairs);

    // Causal GQA attention
    gqa_attn_wmma<<<dim3(S / 16, 32, B), wave, 0, stream>>>(Qb, Kb, Vb, Ab, S);

    // Output projection: out = attn @ Wo^T
    gemm_nt_f32_wmma<<<dim3(2048 / 64, M / 64), wave, 0, stream>>>(Ab, Wo, out, M, 2048, D);
}